// AdvancedGNNModel_40802189312043
// MI455X (gfx1250) — compile-verified
//
#include <hip/hip_runtime.h>
#include <hip/hip_bf16.h>
#include <math.h>

typedef __attribute__((ext_vector_type(2))) float v2f;
typedef __attribute__((ext_vector_type(8))) float v8f;

#define NA_   50000
#define NP__  100000
#define EW_   400000
#define EC_   800000

// ---------- ordered-float <-> uint helpers (monotone map for atomicMax) ----------
__device__ __forceinline__ unsigned f2o(float f) {
    unsigned u = __float_as_uint(f);
    return (u & 0x80000000u) ? ~u : (u | 0x80000000u);
}
__device__ __forceinline__ float o2f(unsigned u) {
    u = (u & 0x80000000u) ? (u & 0x7FFFFFFFu) : ~u;
    return __uint_as_float(u);
}

// =====================================================================
// WMMA fp32 GEMM:  C[M,128] = A[M,K] @ B[K,128] (+bias) (+=C) (ReLU)
// 256 threads = 8 wave32; block tile 64x128; K chunked by 32 through LDS.
// Uses V_WMMA_F32_16X16X4_F32.
// =====================================================================
template <bool ACCUM, bool RELU>
__global__ __launch_bounds__(256)
void gemm128_wmma(const float* __restrict__ A, const float* __restrict__ B,
                  const float* __restrict__ bias, float* __restrict__ C,
                  int M, int K)
{
    __shared__ float As[64][33];    // padded: conflict-free column reads
    __shared__ float Bs[32][128];

    const int tid  = threadIdx.x;
    const int wave = tid >> 5;      // 0..7 -> N tile
    const int lane = tid & 31;
    const int half = lane >> 4;     // 0/1
    const int lq   = lane & 15;
    const int m0   = blockIdx.x * 64;
    const int nt   = wave;

    v8f acc[4] = {};                // 4 M-tiles of 16x16 f32

    for (int k0 = 0; k0 < K; k0 += 32) {
        // stage A 64x32 (zero-pad rows past M)
        #pragma unroll
        for (int i = 0; i < 8; ++i) {
            int idx = tid + 256 * i;          // 0..2047
            int r = idx >> 5, c = idx & 31;
            int gm = m0 + r;
            As[r][c] = (gm < M) ? A[(size_t)gm * K + (k0 + c)] : 0.0f;
        }
        // stage B 32x128
        #pragma unroll
        for (int i = 0; i < 16; ++i) {
            int idx = tid + 256 * i;          // 0..4095
            int r = idx >> 7, c = idx & 127;
            Bs[r][c] = B[(size_t)(k0 + r) * 128 + c];
        }
        __syncthreads();

        #pragma unroll
        for (int kk = 0; kk < 32; kk += 4) {
            const int kb = kk + 2 * half;     // K pair base per half-wave
            v2f bfrag;
            bfrag.x = Bs[kb][nt * 16 + lq];
            bfrag.y = Bs[kb + 1][nt * 16 + lq];
            #pragma unroll
            for (int mt = 0; mt < 4; ++mt) {
                v2f afrag;
                afrag.x = As[mt * 16 + lq][kb];
                afrag.y = As[mt * 16 + lq][kb + 1];
                acc[mt] = __builtin_amdgcn_wmma_f32_16x16x4_f32(
                    false, afrag, false, bfrag, (short)0, acc[mt], false, false);
            }
        }
        __syncthreads();
    }

    // epilogue: C layout -> lane lq = N, vgpr r -> M = r + 8*half
    const int n  = nt * 16 + lq;
    const float bv = bias ? bias[n] : 0.0f;
    #pragma unroll
    for (int mt = 0; mt < 4; ++mt) {
        #pragma unroll
        for (int r = 0; r < 8; ++r) {
            int gm = m0 + mt * 16 + r + 8 * half;
            if (gm < M) {
                size_t off = (size_t)gm * 128 + n;
                float v = acc[mt][r] + bv;
                if (ACCUM) v += C[off];
                if (RELU)  v = fmaxf(v, 0.0f);
                C[off] = v;
            }
        }
    }
}

// ---------- al[n,h] = sum_c hs[n,h*32+c] * a[h*32+c]  (128 thr/node) ----------
__global__ __launch_bounds__(128)
void head_dot(const float* __restrict__ h, const float* __restrict__ avec,
              float* __restrict__ al, int N)
{
    const int n = blockIdx.x;
    const int t = threadIdx.x;        // warp = head (wave32)
    float p = h[(size_t)n * 128 + t] * avec[t];
    #pragma unroll
    for (int o = 16; o >= 1; o >>= 1) p += __shfl_xor(p, o, 32);
    if ((t & 31) == 0) al[n * 4 + (t >> 5)] = p;
}

// ---------- out[n,c] = bias[c] ----------
__global__ void bias_init(float* __restrict__ out, const float* __restrict__ bias, int total)
{
    int tid = blockIdx.x * blockDim.x + threadIdx.x;
    if (tid < total) out[tid] = bias[tid & 127];
}

// ---------- GAT edge pass 1: segment max of leaky_relu(al_s[src]+al_d[dst]) ----------
__global__ void gat_edge_max(const int* __restrict__ src, const int* __restrict__ dst,
                             const float* __restrict__ als, const float* __restrict__ ald,
                             unsigned* __restrict__ m, int E)
{
    int e = blockIdx.x * blockDim.x + threadIdx.x;
    if (e >= E) return;
    int s = src[e], d = dst[e];
    #pragma unroll
    for (int h = 0; h < 4; ++h) {
        float x = als[s * 4 + h] + ald[d * 4 + h];
        x = x > 0.0f ? x : 0.2f * x;
        atomicMax(&m[d * 4 + h], f2o(x));
    }
}

// ---------- GAT edge pass 2: ex = exp(e - max); segment sum ----------
__global__ void gat_edge_expsum(const int* __restrict__ src, const int* __restrict__ dst,
                                const float* __restrict__ als, const float* __restrict__ ald,
                                const unsigned* __restrict__ m, float* __restrict__ ex,
                                float* __restrict__ ssum, int E)
{
    int e = blockIdx.x * blockDim.x + threadIdx.x;
    if (e >= E) return;
    int s = src[e], d = dst[e];
    #pragma unroll
    for (int h = 0; h < 4; ++h) {
        float x = als[s * 4 + h] + ald[d * 4 + h];
        x = x > 0.0f ? x : 0.2f * x;
        float ev = expf(x - o2f(m[d * 4 + h]));
        ex[(size_t)e * 4 + h] = ev;
        atomicAdd(&ssum[d * 4 + h], ev);
    }
}

// ---------- GAT edge pass 3: out[dst] += alpha * hs[src]  (128 thr/edge) ----------
__global__ void gat_edge_msg(const int* __restrict__ src, const int* __restrict__ dst,
                             const float* __restrict__ ex, const float* __restrict__ ssum,
                             const float* __restrict__ hs, float* __restrict__ out, int E)
{
    int tid = blockIdx.x * blockDim.x + threadIdx.x;
    int e = tid >> 7;
    if (e >= E) return;
    int c = tid & 127, h = c >> 5;
    int s = src[e], d = dst[e];
    float alpha = ex[(size_t)e * 4 + h] / (ssum[d * 4 + h] + 1e-16f);
    atomicAdd(&out[(size_t)d * 128 + c], alpha * hs[(size_t)s * 128 + c]);
}

// ---------- SAGE: agg[dst] += h[src]; cnt[dst] += 1 ----------
__global__ void sage_agg(const int* __restrict__ src, const int* __restrict__ dst,
                         const float* __restrict__ hp, float* __restrict__ agg,
                         float* __restrict__ cnt, int E)
{
    int tid = blockIdx.x * blockDim.x + threadIdx.x;
    int e = tid >> 7;
    if (e >= E) return;
    int c = tid & 127;
    int s = src[e], d = dst[e];
    atomicAdd(&agg[(size_t)d * 128 + c], hp[(size_t)s * 128 + c]);
    if (c == 0) atomicAdd(&cnt[d], 1.0f);
}

__global__ void sage_mean(float* __restrict__ agg, const float* __restrict__ cnt, int N)
{
    int tid = blockIdx.x * blockDim.x + threadIdx.x;
    int n = tid >> 7;
    if (n >= N) return;
    agg[tid] /= fmaxf(cnt[n], 1.0f);
}

// ---------- h = relu(layernorm(out)*g+b) + h  (128 thr/row) ----------
__global__ __launch_bounds__(128)
void ln_relu_res(const float* __restrict__ out, const float* __restrict__ g,
                 const float* __restrict__ b, float* __restrict__ h, int N)
{
    const int n = blockIdx.x, t = threadIdx.x;
    __shared__ float s1[4], s2[4];
    float v = out[(size_t)n * 128 + t];
    float p = v, q = v * v;
    #pragma unroll
    for (int o = 16; o >= 1; o >>= 1) { p += __shfl_xor(p, o, 32); q += __shfl_xor(q, o, 32); }
    if ((t & 31) == 0) { s1[t >> 5] = p; s2[t >> 5] = q; }
    __syncthreads();
    float S1 = s1[0] + s1[1] + s1[2] + s1[3];
    float S2 = s2[0] + s2[1] + s2[2] + s2[3];
    float mu  = S1 * (1.0f / 128.0f);
    float var = S2 * (1.0f / 128.0f) - mu * mu;
    float y = (v - mu) * rsqrtf(var + 1e-5f) * g[t] + b[t];
    h[(size_t)n * 128 + t] = fmaxf(y, 0.0f) + h[(size_t)n * 128 + t];
}

// =====================================================================
static inline void launch_gemm(const float* A, const float* B, const float* bias, float* C,
                               int M, int K, bool accum, bool relu, hipStream_t s)
{
    dim3 g((M + 63) / 64), blk(256);
    if (!accum && relu)       gemm128_wmma<false, true ><<<g, blk, 0, s>>>(A, B, bias, C, M, K);
    else if (!accum && !relu) gemm128_wmma<false, false><<<g, blk, 0, s>>>(A, B, bias, C, M, K);
    else if (accum && relu)   gemm128_wmma<true,  true ><<<g, blk, 0, s>>>(A, B, bias, C, M, K);
    else                      gemm128_wmma<true,  false><<<g, blk, 0, s>>>(A, B, bias, C, M, K);
}

extern "C" void kernel_launch(void* const* d_in, const int* in_sizes, int n_in,
                              void* d_out, int out_size, void* d_ws, size_t ws_size,
                              hipStream_t stream)
{
    // ---- inputs (setup_inputs order) ----
    const float* x_author       = (const float*)d_in[0];
    const float* x_paper        = (const float*)d_in[1];
    const float* emb_author_W   = (const float*)d_in[2];
    const float* emb_author_b   = (const float*)d_in[3];
    const float* emb_paper_W    = (const float*)d_in[4];
    const float* emb_paper_b    = (const float*)d_in[5];
    const float* gatw_Wsrc      = (const float*)d_in[6];
    const float* gatw_Wdst      = (const float*)d_in[7];
    const float* gatw_asrc      = (const float*)d_in[8];
    const float* gatw_adst      = (const float*)d_in[9];
    const float* gatw_b         = (const float*)d_in[10];
    const float* gata_Wsrc      = (const float*)d_in[11];
    const float* gata_Wdst      = (const float*)d_in[12];
    const float* gata_asrc      = (const float*)d_in[13];
    const float* gata_adst      = (const float*)d_in[14];
    const float* gata_b         = (const float*)d_in[15];
    const float* sage_Wl        = (const float*)d_in[16];
    const float* sage_bl        = (const float*)d_in[17];
    const float* sage_Wr        = (const float*)d_in[18];
    const float* sage_br        = (const float*)d_in[19];
    const float* ln_a_g         = (const float*)d_in[20];
    const float* ln_a_b         = (const float*)d_in[21];
    const float* ln_p_g         = (const float*)d_in[22];
    const float* ln_p_b         = (const float*)d_in[23];
    const float* out_author_W   = (const float*)d_in[24];
    const float* out_author_b   = (const float*)d_in[25];
    const float* out_paper_W    = (const float*)d_in[26];
    const float* out_paper_b    = (const float*)d_in[27];
    const int* writes_src = (const int*)d_in[28];
    const int* writes_dst = (const int*)d_in[29];
    const int* cites_src  = (const int*)d_in[30];
    const int* cites_dst  = (const int*)d_in[31];
    const int* auth_src   = (const int*)d_in[32];
    const int* auth_dst   = (const int*)d_in[33];

    // ---- workspace layout (floats) ----
    float* ws = (float*)d_ws;
    size_t o = 0;
    float* h_a   = ws + o; o += (size_t)NA_  * 128;   // author features
    float* h_p   = ws + o; o += (size_t)NP__ * 128;   // paper features
    float* bufA  = ws + o; o += (size_t)NA_  * 128;   // hs_w then hd_a
    float* bufP  = ws + o; o += (size_t)NP__ * 128;   // hd_w then hs_a then SAGE agg
    float* out_p = ws + o; o += (size_t)NP__ * 128;
    float* out_a = ws + o; o += (size_t)NA_  * 128;
    float* al_s_w = ws + o; o += (size_t)NA_  * 4;
    float* al_d_w = ws + o; o += (size_t)NP__ * 4;
    float* al_s_a = ws + o; o += (size_t)NP__ * 4;
    float* al_d_a = ws + o; o += (size_t)NA_  * 4;
    unsigned* m_w = (unsigned*)(ws + o); o += (size_t)NP__ * 4;
    float*    s_w = ws + o;              o += (size_t)NP__ * 4;
    unsigned* m_a = (unsigned*)(ws + o); o += (size_t)NA_  * 4;
    float*    s_a = ws + o;              o += (size_t)NA_  * 4;
    float* ex_w = ws + o; o += (size_t)EW_ * 4;
    float* ex_a = ws + o; o += (size_t)EW_ * 4;
    float* cnt  = ws + o; o += (size_t)NP__;

    const int B256 = 256;
    auto blocks = [](long long n, int b) { return (unsigned)((n + b - 1) / b); };

    // ---- input embeddings + ReLU ----
    launch_gemm(x_author, emb_author_W, emb_author_b, h_a, NA_, 128, false, true, stream);
    launch_gemm(x_paper,  emb_paper_W,  emb_paper_b,  h_p, NP__, 256, false, true, stream);

    for (int l = 0; l < 2; ++l) {
        const size_t Wo = (size_t)l * 128 * 128, vo = (size_t)l * 128;

        // ===== writes GAT (author -> paper) =====
        launch_gemm(h_a, gatw_Wsrc + Wo, nullptr, bufA, NA_, 128, false, false, stream);  // hs_w
        launch_gemm(h_p, gatw_Wdst + Wo, nullptr, bufP, NP__, 128, false, false, stream); // hd_w
        head_dot<<<NA_, 128, 0, stream>>>(bufA, gatw_asrc + vo, al_s_w, NA_);
        head_dot<<<NP__, 128, 0, stream>>>(bufP, gatw_adst + vo, al_d_w, NP__);
        hipMemsetAsync(m_w, 0, (size_t)NP__ * 4 * sizeof(unsigned), stream);
        hipMemsetAsync(s_w, 0, (size_t)NP__ * 4 * sizeof(float), stream);
        bias_init<<<blocks((long long)NP__ * 128, B256), B256, 0, stream>>>(out_p, gatw_b + vo, NP__ * 128);
        gat_edge_max<<<blocks(EW_, B256), B256, 0, stream>>>(writes_src, writes_dst, al_s_w, al_d_w, m_w, EW_);
        gat_edge_expsum<<<blocks(EW_, B256), B256, 0, stream>>>(writes_src, writes_dst, al_s_w, al_d_w, m_w, ex_w, s_w, EW_);
        gat_edge_msg<<<blocks((long long)EW_ * 128, B256), B256, 0, stream>>>(writes_src, writes_dst, ex_w, s_w, bufA, out_p, EW_);

        // ===== auth GAT (paper -> author) ===== (reuse bufP as hs_a, bufA as hd_a)
        launch_gemm(h_p, gata_Wsrc + Wo, nullptr, bufP, NP__, 128, false, false, stream); // hs_a
        launch_gemm(h_a, gata_Wdst + Wo, nullptr, bufA, NA_, 128, false, false, stream);  // hd_a
        head_dot<<<NP__, 128, 0, stream>>>(bufP, gata_asrc + vo, al_s_a, NP__);
        head_dot<<<NA_, 128, 0, stream>>>(bufA, gata_adst + vo, al_d_a, NA_);
        hipMemsetAsync(m_a, 0, (size_t)NA_ * 4 * sizeof(unsigned), stream);
        hipMemsetAsync(s_a, 0, (size_t)NA_ * 4 * sizeof(float), stream);
        bias_init<<<blocks((long long)NA_ * 128, B256), B256, 0, stream>>>(out_a, gata_b + vo, NA_ * 128);
        gat_edge_max<<<blocks(EW_, B256), B256, 0, stream>>>(auth_src, auth_dst, al_s_a, al_d_a, m_a, EW_);
        gat_edge_expsum<<<blocks(EW_, B256), B256, 0, stream>>>(auth_src, auth_dst, al_s_a, al_d_a, m_a, ex_a, s_a, EW_);
        gat_edge_msg<<<blocks((long long)EW_ * 128, B256), B256, 0, stream>>>(auth_src, auth_dst, ex_a, s_a, bufP, out_a, EW_);

        // ===== SAGE cites (paper -> paper), summed into out_p ===== (reuse bufP as agg)
        hipMemsetAsync(bufP, 0, (size_t)NP__ * 128 * sizeof(float), stream);
        hipMemsetAsync(cnt, 0, (size_t)NP__ * sizeof(float), stream);
        sage_agg<<<blocks((long long)EC_ * 128, B256), B256, 0, stream>>>(cites_src, cites_dst, h_p, bufP, cnt, EC_);
        sage_mean<<<blocks((long long)NP__ * 128, B256), B256, 0, stream>>>(bufP, cnt, NP__);
        launch_gemm(bufP, sage_Wl + Wo, sage_bl + vo, out_p, NP__, 128, true, false, stream);
        launch_gemm(h_p,  sage_Wr + Wo, sage_br + vo, out_p, NP__, 128, true, false, stream);

        // ===== LayerNorm -> ReLU -> residual (in place) =====
        ln_relu_res<<<NP__, 128, 0, stream>>>(out_p, ln_p_g + vo, ln_p_b + vo, h_p, NP__);
        ln_relu_res<<<NA_, 128, 0, stream>>>(out_a, ln_a_g + vo, ln_a_b + vo, h_a, NA_);
    }

    // ---- output heads: concat (author, paper) ----
    float* y = (float*)d_out;
    launch_gemm(h_a, out_author_W, out_author_b, y, NA_, 128, false, false, stream);
    launch_gemm(h_p, out_paper_W, out_paper_b, y + (size_t)NA_ * 128, NP__, 128, false, false, stream);
}